// GCN_Plus_N2V_1984274891426
// MI455X (gfx1250) — compile-verified
//
#include <hip/hip_runtime.h>
#include <math.h>

typedef __attribute__((ext_vector_type(2))) float v2f;
typedef __attribute__((ext_vector_type(8))) float v8f;

#define FIN  128
#define EMBD 64
#define DIN  192   // FIN + EMBD
#define HDIM 64
#define CLS  16

// D = A(16x4) * B(4x16) + C  -- f32 WMMA
__device__ __forceinline__ v8f wmma_f32(v2f a, v2f b, v8f c) {
    return __builtin_amdgcn_wmma_f32_16x16x4_f32(false, a, false, b, (short)0, c, false, false);
}

// Stage W[K x NC] (row-major) into LDS with K-pairs interleaved:
//   sP[(k>>1)*(2*NC) + n*2 + (k&1)] = W[k*NC + n]
// so the WMMA B-fragment (W[ka][n], W[ka+1][n]) is one aligned 8-byte word.
template <int K, int NC>
__device__ __forceinline__ void stage_w_pairs(const float* __restrict__ W, float* sP) {
    for (int i = threadIdx.x; i < K * NC; i += 256) {
        int k = i / NC, n = i % NC;
        sP[(k >> 1) * (2 * NC) + (n << 1) + (k & 1)] = W[i];
    }
}

// ---------------- degree / norm ----------------
__global__ void k_init_deg(float* deg, int N) {
    int i = blockIdx.x * blockDim.x + threadIdx.x;
    if (i < N) deg[i] = 1.0f;                  // self-loop weight 1
}

__global__ void k_accum_deg(const long long* __restrict__ eidx,
                            const float* __restrict__ w, float* deg, int E) {
    int e = blockIdx.x * blockDim.x + threadIdx.x;
    if (e < E) {
        long long d = __builtin_nontemporal_load(eidx + (size_t)E + e);
        float     wv = __builtin_nontemporal_load(w + e);
        atomicAdd(deg + d, wv);
    }
}

__global__ void k_dinv(float* deg, int N) {
    int i = blockIdx.x * blockDim.x + threadIdx.x;
    if (i < N) { float v = deg[i]; deg[i] = (v > 0.0f) ? rsqrtf(v) : 0.0f; }
}

// ---------------- GEMM1: relu([x|n2v] @ W_first + b) ----------------
__global__ __launch_bounds__(256)
void k_gemm_first(const float* __restrict__ x, const float* __restrict__ emb,
                  const float* __restrict__ W, const float* __restrict__ b,
                  float* __restrict__ out, int N) {
    __shared__ float sP[DIN * HDIM];           // 48 KB, pair-interleaved
    stage_w_pairs<DIN, HDIM>(W, sP);
    __syncthreads();

    int wave = threadIdx.x >> 5, lane = threadIdx.x & 31;
    int tile = blockIdx.x * 8 + wave;
    int ntiles = (N + 15) >> 4;
    if (tile >= ntiles) return;

    int mrow = tile * 16 + (lane & 15);
    if (mrow >= N) mrow = N - 1;               // clamp: keep EXEC all-ones for WMMA
    int khalf = (lane >> 4) * 2;               // 0 or 2 (even)
    int ncol  = lane & 15;

    v8f acc0 = {}, acc1 = {}, acc2 = {}, acc3 = {};
    for (int k = 0; k < DIN; k += 4) {
        int ka = k + khalf;                    // even
        v2f a;
        if (ka < FIN) a = *(const v2f*)(x   + (size_t)mrow * FIN  + ka);
        else          a = *(const v2f*)(emb + (size_t)mrow * EMBD + (ka - FIN));
        const float* wp = sP + (size_t)(ka >> 1) * (2 * HDIM) + (ncol << 1);
        v2f b0 = *(const v2f*)(wp);
        v2f b1 = *(const v2f*)(wp + 32);
        v2f b2 = *(const v2f*)(wp + 64);
        v2f b3 = *(const v2f*)(wp + 96);
        acc0 = wmma_f32(a, b0, acc0);
        acc1 = wmma_f32(a, b1, acc1);
        acc2 = wmma_f32(a, b2, acc2);
        acc3 = wmma_f32(a, b3, acc3);
    }
    int half = lane >> 4;
    #pragma unroll
    for (int v = 0; v < 8; ++v) {
        int row = tile * 16 + v + 8 * half;
        if (row < N) {
            float* o = out + (size_t)row * HDIM + ncol;
            o[0]  = fmaxf(acc0[v] + b[ncol],      0.0f);
            o[16] = fmaxf(acc1[v] + b[16 + ncol], 0.0f);
            o[32] = fmaxf(acc2[v] + b[32 + ncol], 0.0f);
            o[48] = fmaxf(acc3[v] + b[48 + ncol], 0.0f);
        }
    }
}

// ---------------- GEMM: t = h @ W  (64x64, no bias/act) ----------------
__global__ __launch_bounds__(256)
void k_gemm_h(const float* __restrict__ hin, const float* __restrict__ W,
              float* __restrict__ out, int N) {
    __shared__ float sP[HDIM * HDIM];          // 16 KB, pair-interleaved
    stage_w_pairs<HDIM, HDIM>(W, sP);
    __syncthreads();

    int wave = threadIdx.x >> 5, lane = threadIdx.x & 31;
    int tile = blockIdx.x * 8 + wave;
    int ntiles = (N + 15) >> 4;
    if (tile >= ntiles) return;

    int mrow = tile * 16 + (lane & 15);
    if (mrow >= N) mrow = N - 1;
    int khalf = (lane >> 4) * 2;
    int ncol  = lane & 15;

    v8f acc0 = {}, acc1 = {}, acc2 = {}, acc3 = {};
    for (int k = 0; k < HDIM; k += 4) {
        int ka = k + khalf;
        v2f a = *(const v2f*)(hin + (size_t)mrow * HDIM + ka);
        const float* wp = sP + (size_t)(ka >> 1) * (2 * HDIM) + (ncol << 1);
        v2f b0 = *(const v2f*)(wp);
        v2f b1 = *(const v2f*)(wp + 32);
        v2f b2 = *(const v2f*)(wp + 64);
        v2f b3 = *(const v2f*)(wp + 96);
        acc0 = wmma_f32(a, b0, acc0);
        acc1 = wmma_f32(a, b1, acc1);
        acc2 = wmma_f32(a, b2, acc2);
        acc3 = wmma_f32(a, b3, acc3);
    }
    int half = lane >> 4;
    #pragma unroll
    for (int v = 0; v < 8; ++v) {
        int row = tile * 16 + v + 8 * half;
        if (row < N) {
            float* o = out + (size_t)row * HDIM + ncol;
            o[0] = acc0[v]; o[16] = acc1[v]; o[32] = acc2[v]; o[48] = acc3[v];
        }
    }
}

// ---------------- self-loop init: agg = t * dinv^2 ----------------
__global__ void k_selfinit(const float* __restrict__ t, const float* __restrict__ dinv,
                           float* __restrict__ agg, int N) {
    size_t idx = (size_t)blockIdx.x * blockDim.x + threadIdx.x;
    if (idx < (size_t)N * HDIM) {
        float di = dinv[idx >> 6];
        agg[idx] = t[idx] * di * di;
    }
}

// ---------------- edge scatter: one wave per edge, 2 feats/lane ----------------
__global__ __launch_bounds__(256)
void k_scatter(const float* __restrict__ t, const long long* __restrict__ eidx,
               const float* __restrict__ w, const float* __restrict__ dinv,
               float* __restrict__ agg, int E) {
    unsigned gid = blockIdx.x * 256u + threadIdx.x;
    unsigned e = gid >> 5;
    if (e >= (unsigned)E) return;
    int lane = threadIdx.x & 31;
    // wave-uniform edge metadata; NT: stream past L2 to keep node tables resident
    size_t s = (size_t)__builtin_nontemporal_load(eidx + e);
    size_t d = (size_t)__builtin_nontemporal_load(eidx + (size_t)E + e);
    float  c = dinv[s] * __builtin_nontemporal_load(w + e) * dinv[d];
    const float2 tv = *(const float2*)(t + s * HDIM + lane * 2);   // L2-resident gather
    atomicAdd(agg + d * HDIM + lane * 2,     tv.x * c);
    atomicAdd(agg + d * HDIM + lane * 2 + 1, tv.y * c);
}

// ---------------- bias + relu ----------------
__global__ void k_bias_relu(const float* __restrict__ agg, const float* __restrict__ b,
                            float* __restrict__ hout, int N) {
    size_t idx = (size_t)blockIdx.x * blockDim.x + threadIdx.x;
    if (idx < (size_t)N * HDIM)
        hout[idx] = fmaxf(agg[idx] + b[idx & (HDIM - 1)], 0.0f);
}

// ---------------- classifier GEMM + fused log_softmax ----------------
__global__ __launch_bounds__(256)
void k_out_lsm(const float* __restrict__ hin, const float* __restrict__ W,
               const float* __restrict__ b, float* __restrict__ out, int N) {
    __shared__ float sP[HDIM * CLS];           // 4 KB, pair-interleaved
    stage_w_pairs<HDIM, CLS>(W, sP);
    __syncthreads();

    int wave = threadIdx.x >> 5, lane = threadIdx.x & 31;
    int tile = blockIdx.x * 8 + wave;
    int ntiles = (N + 15) >> 4;
    if (tile >= ntiles) return;

    int mrow = tile * 16 + (lane & 15);
    if (mrow >= N) mrow = N - 1;
    int khalf = (lane >> 4) * 2;
    int ncol  = lane & 15;

    v8f acc = {};
    for (int k = 0; k < HDIM; k += 4) {
        int ka = k + khalf;
        v2f a = *(const v2f*)(hin + (size_t)mrow * HDIM + ka);
        v2f bb = *(const v2f*)(sP + (size_t)(ka >> 1) * (2 * CLS) + (ncol << 1));
        acc = wmma_f32(a, bb, acc);
    }
    float bias = b[ncol];
    int half = lane >> 4;
    #pragma unroll
    for (int v = 0; v < 8; ++v) {
        float r = acc[v] + bias;
        // row of 16 logits lives across lanes 0..15 of this half-wave
        float m = r;
        m = fmaxf(m, __shfl_xor(m, 1, 32));
        m = fmaxf(m, __shfl_xor(m, 2, 32));
        m = fmaxf(m, __shfl_xor(m, 4, 32));
        m = fmaxf(m, __shfl_xor(m, 8, 32));
        float s = __expf(r - m);
        s += __shfl_xor(s, 1, 32);
        s += __shfl_xor(s, 2, 32);
        s += __shfl_xor(s, 4, 32);
        s += __shfl_xor(s, 8, 32);
        float res = (r - m) - __logf(s);
        int row = tile * 16 + v + 8 * half;
        if (row < N) out[(size_t)row * CLS + ncol] = res;
    }
}

extern "C" void kernel_launch(void* const* d_in, const int* in_sizes, int n_in,
                              void* d_out, int out_size, void* d_ws, size_t ws_size,
                              hipStream_t stream) {
    const float*     x    = (const float*)d_in[0];
    const long long* eidx = (const long long*)d_in[1];
    const float*     ew   = (const float*)d_in[2];
    const float*     emb  = (const float*)d_in[3];
    const float*     Wf   = (const float*)d_in[4];
    const float*     bf   = (const float*)d_in[5];
    const float*     W1   = (const float*)d_in[6];
    const float*     b1   = (const float*)d_in[7];
    const float*     W2   = (const float*)d_in[8];
    const float*     b2   = (const float*)d_in[9];
    const float*     Wo   = (const float*)d_in[10];
    const float*     bo   = (const float*)d_in[11];
    float*           out  = (float*)d_out;

    const int N = in_sizes[0] / FIN;
    const int E = in_sizes[2];

    // workspace layout (all L2-resident: ~77 MB vs 192 MB L2)
    float* deg = (float*)d_ws;                                 // N (then holds dinv)
    float* h   = deg + (((size_t)N + 63) & ~(size_t)63);       // N*64
    float* t   = h   + (size_t)N * HDIM;                       // N*64
    float* agg = t   + (size_t)N * HDIM;                       // N*64

    const int nBlk  = (N + 255) / 256;
    const int eBlk  = (E + 255) / 256;
    const int fBlk  = (int)(((size_t)N * HDIM + 255) / 256);
    const int gBlk  = (((N + 15) / 16) + 7) / 8;               // 8 wave-tiles per block
    const int sBlk  = (int)((((size_t)E << 5) + 255) / 256);   // one wave32 per edge

    // GCN normalization
    k_init_deg <<<nBlk, 256, 0, stream>>>(deg, N);
    k_accum_deg<<<eBlk, 256, 0, stream>>>(eidx, ew, deg, E);
    k_dinv     <<<nBlk, 256, 0, stream>>>(deg, N);

    // first transform
    k_gemm_first<<<gBlk, 256, 0, stream>>>(x, emb, Wf, bf, h, N);

    // conv layer 1
    k_gemm_h   <<<gBlk, 256, 0, stream>>>(h, W1, t, N);
    k_selfinit <<<fBlk, 256, 0, stream>>>(t, deg, agg, N);
    k_scatter  <<<sBlk, 256, 0, stream>>>(t, eidx, ew, deg, agg, E);
    k_bias_relu<<<fBlk, 256, 0, stream>>>(agg, b1, h, N);

    // conv layer 2
    k_gemm_h   <<<gBlk, 256, 0, stream>>>(h, W2, t, N);
    k_selfinit <<<fBlk, 256, 0, stream>>>(t, deg, agg, N);
    k_scatter  <<<sBlk, 256, 0, stream>>>(t, eidx, ew, deg, agg, E);
    k_bias_relu<<<fBlk, 256, 0, stream>>>(agg, b2, h, N);

    // classifier + log_softmax
    k_out_lsm  <<<gBlk, 256, 0, stream>>>(h, Wo, bo, out, N);
}